// LocationAwareAttention_35330400977295
// MI455X (gfx1250) — compile-verified
//
#include <hip/hip_runtime.h>
#include <hip/hip_bf16.h>
#include <math.h>

#define B_SZ 256
#define L_SZ 1024
#define RNN_DIM 1024
#define A_DIM 128
#define ENC_DIM 512
#define NF 32
#define KER 31

typedef float v2f __attribute__((ext_vector_type(2)));
typedef float v8f __attribute__((ext_vector_type(8)));

#if defined(__AMDGCN__) && __has_builtin(__builtin_amdgcn_wmma_f32_16x16x4_f32)
#define HAVE_WMMA_F32 1
#else
#define HAVE_WMMA_F32 0
#endif

// ---------------------------------------------------------------------------
// Kernel 1: pq = query @ W_query^T  (256x128, K=1024) via fp32 WMMA 16x16x4.
// One wave per 16x16 output tile. grid = (ntiles=8, mtiles=16), block = 32.
// ---------------------------------------------------------------------------
__global__ __launch_bounds__(32) void pq_wmma_kernel(
    const float* __restrict__ q,    // (256, 1024)
    const float* __restrict__ Wq,   // (128, 1024)
    float* __restrict__ pq)         // (256, 128)
{
  const int nt   = blockIdx.x;          // 0..7
  const int mt   = blockIdx.y;          // 0..15
  const int lane = threadIdx.x;
  const int half = lane >> 4;           // 0: K=k,k+1   1: K=k+2,k+3
  const int l16  = lane & 15;
  const int row  = mt * 16 + l16;       // A-matrix row (query row)
  const int col  = nt * 16 + l16;       // B-matrix column (W_query row)
#if HAVE_WMMA_F32
  const float* qrow = q  + (size_t)row * RNN_DIM + half * 2;
  const float* wrow = Wq + (size_t)col * RNN_DIM + half * 2;
  v8f c = {};
  for (int k = 0; k < RNN_DIM; k += 4) {
    v2f a, b;
    a[0] = qrow[k];  a[1] = qrow[k + 1];      // A 16x4 fp32 layout
    b[0] = wrow[k];  b[1] = wrow[k + 1];      // B 4x16 fp32 layout (B[k][n]=Wq[n][k])
    c = __builtin_amdgcn_wmma_f32_16x16x4_f32(
        /*neg_a=*/false, a, /*neg_b=*/false, b,
        /*c_mod=*/(short)0, c, /*reuse_a=*/false, /*reuse_b=*/false);
  }
  #pragma unroll
  for (int r = 0; r < 8; ++r) {
    int m = r + half * 8;                      // D 16x16 fp32 layout
    pq[(size_t)(mt * 16 + m) * A_DIM + (nt * 16 + l16)] = c[r];
  }
#else
  // Scalar fallback with the same output mapping.
  #pragma unroll 1
  for (int r = 0; r < 8; ++r) {
    int m = r + half * 8;
    const float* qr = q  + (size_t)(mt * 16 + m) * RNN_DIM;
    const float* wr = Wq + (size_t)col * RNN_DIM;
    float acc = 0.f;
    for (int k = 0; k < RNN_DIM; ++k) acc = fmaf(qr[k], wr[k], acc);
    pq[(size_t)(mt * 16 + m) * A_DIM + (nt * 16 + l16)] = acc;
  }
#endif
}

// ---------------------------------------------------------------------------
// Kernel 2: fused conv1d + location dense + tanh + energy dot.
// One wave per l; block = 8 waves covers 64 l's; grid = (L/64, B).
// ---------------------------------------------------------------------------
__global__ __launch_bounds__(256) void energies_kernel(
    const float* __restrict__ ps,        // (B, L, 128)
    const float* __restrict__ attn,      // (B, 2, L)
    const unsigned char* __restrict__ mask, // (B, L) bool
    const float* __restrict__ pq,        // (B, 128)
    const float* __restrict__ Wc,        // (32, 2, 31)
    const float* __restrict__ Wl,        // (128, 32)
    const float* __restrict__ Wv,        // (128)
    const float* __restrict__ bv,        // (1)
    float* __restrict__ energies)        // (B, L)
{
  __shared__ float sWl[NF][A_DIM];       // transposed: [f][a]
  __shared__ float sWc[2 * KER][NF];     // transposed: [c*31+k][f]
  __shared__ float sPq[A_DIM];
  __shared__ float sWv[A_DIM];
  __shared__ float sLocf[8][NF];         // per-wave conv features

  const int b   = blockIdx.y;
  const int tid = threadIdx.x;

  for (int i = tid; i < NF * A_DIM; i += 256) {
    int a = i & (A_DIM - 1), f = i >> 7;
    sWl[f][a] = Wl[a * NF + f];
  }
  for (int i = tid; i < 2 * KER * NF; i += 256) {
    int f = i & (NF - 1), ck = i >> 5;
    sWc[ck][f] = Wc[f * (2 * KER) + ck];
  }
  if (tid < A_DIM) { sPq[tid] = pq[b * A_DIM + tid]; sWv[tid] = Wv[tid]; }
  __syncthreads();

  const int wave = tid >> 5, lane = tid & 31;
  const float bvv = bv[0];
  const float* attnb = attn + (size_t)b * 2 * L_SZ;

  for (int it = 0; it < 8; ++it) {
    const int l = blockIdx.x * 64 + wave * 8 + it;
    // conv: lane computes filter f = lane
    float lf = 0.f;
    #pragma unroll
    for (int c = 0; c < 2; ++c) {
      #pragma unroll
      for (int k = 0; k < KER; ++k) {
        int pos = l + k - 15;
        float x = (pos >= 0 && pos < L_SZ) ? attnb[c * L_SZ + pos] : 0.f;
        lf = fmaf(x, sWc[c * KER + k][lane], lf);
      }
    }
    __syncthreads();
    sLocf[wave][lane] = lf;
    __syncthreads();

    const float* psrow = ps + ((size_t)b * L_SZ + l) * A_DIM;
    float part = 0.f;
    #pragma unroll
    for (int j = 0; j < 4; ++j) {
      int a = lane + 32 * j;
      float acc = sPq[a] + psrow[a];
      #pragma unroll
      for (int f = 0; f < NF; ++f) acc = fmaf(sLocf[wave][f], sWl[f][a], acc);
      // tanh(x) = 1 - 2/(exp(2x)+1)  (exact at +-inf, ~ulp-level elsewhere)
      float t = 1.0f - 2.0f / (__expf(2.0f * acc) + 1.0f);
      part = fmaf(sWv[a], t, part);
    }
    #pragma unroll
    for (int off = 16; off > 0; off >>= 1) part += __shfl_xor(part, off, 32);
    if (lane == 0) {
      float e = part + bvv;
      if (!mask[(size_t)b * L_SZ + l]) e = -INFINITY;
      energies[(size_t)b * L_SZ + l] = e;
    }
  }
}

// ---------------------------------------------------------------------------
// Kernel 3: row softmax over L=1024. One block (256 threads) per batch row.
// ---------------------------------------------------------------------------
__global__ __launch_bounds__(256) void softmax_kernel(
    const float* __restrict__ e_in, float* __restrict__ w_out)
{
  __shared__ float red[256];
  const int b = blockIdx.x, t = threadIdx.x;
  const float* e = e_in + (size_t)b * L_SZ;
  float v0 = e[t], v1 = e[t + 256], v2 = e[t + 512], v3 = e[t + 768];
  float m = fmaxf(fmaxf(v0, v1), fmaxf(v2, v3));
  red[t] = m; __syncthreads();
  for (int s = 128; s > 0; s >>= 1) {
    if (t < s) red[t] = fmaxf(red[t], red[t + s]);
    __syncthreads();
  }
  m = red[0]; __syncthreads();
  v0 = __expf(v0 - m); v1 = __expf(v1 - m);
  v2 = __expf(v2 - m); v3 = __expf(v3 - m);
  red[t] = (v0 + v1) + (v2 + v3); __syncthreads();
  for (int s = 128; s > 0; s >>= 1) {
    if (t < s) red[t] += red[t + s];
    __syncthreads();
  }
  float inv = 1.0f / red[0];
  float* w = w_out + (size_t)b * L_SZ;
  w[t] = v0 * inv; w[t + 256] = v1 * inv;
  w[t + 512] = v2 * inv; w[t + 768] = v3 * inv;
}

// ---------------------------------------------------------------------------
// Kernel 4: context partials. grid = (4 l-chunks, B), block = 512 (one per d).
// Streams `source` coalesced (2KB rows); weights broadcast from LDS.
// ---------------------------------------------------------------------------
__global__ __launch_bounds__(512) void context_partial_kernel(
    const float* __restrict__ w,      // (B, L) attention weights
    const float* __restrict__ src,    // (B, L, 512)
    float* __restrict__ partial)      // (B, 4, 512)
{
  __shared__ float sw[256];
  const int b = blockIdx.y, chunk = blockIdx.x, d = threadIdx.x;
  if (threadIdx.x < 256)
    sw[threadIdx.x] = w[(size_t)b * L_SZ + chunk * 256 + threadIdx.x];
  __syncthreads();
  const float* s = src + ((size_t)b * L_SZ + (size_t)chunk * 256) * ENC_DIM + d;
  float acc0 = 0.f, acc1 = 0.f;
  #pragma unroll 2
  for (int i = 0; i < 256; i += 2) {
    if (i + 32 < 256)
      __builtin_prefetch(&s[(size_t)(i + 32) * ENC_DIM], 0, 0);  // global_prefetch_b8
    acc0 = fmaf(sw[i],     s[(size_t)i * ENC_DIM],        acc0);
    acc1 = fmaf(sw[i + 1], s[(size_t)(i + 1) * ENC_DIM],  acc1);
  }
  partial[((size_t)b * 4 + chunk) * ENC_DIM + d] = acc0 + acc1;
}

__global__ __launch_bounds__(256) void context_reduce_kernel(
    const float* __restrict__ partial, float* __restrict__ ctx)
{
  int i = blockIdx.x * 256 + threadIdx.x;   // 0 .. 256*512-1
  if (i < B_SZ * ENC_DIM) {
    int b = i >> 9, d = i & (ENC_DIM - 1);
    const float* p = partial + (size_t)b * 4 * ENC_DIM + d;
    ctx[i] = (p[0] + p[ENC_DIM]) + (p[2 * ENC_DIM] + p[3 * ENC_DIM]);
  }
}

// ---------------------------------------------------------------------------
extern "C" void kernel_launch(void* const* d_in, const int* in_sizes, int n_in,
                              void* d_out, int out_size, void* d_ws, size_t ws_size,
                              hipStream_t stream) {
  const float* query    = (const float*)d_in[0];
  const float* source   = (const float*)d_in[1];
  const float* ps       = (const float*)d_in[2];
  const float* attn_cat = (const float*)d_in[3];
  const unsigned char* mask = (const unsigned char*)d_in[4];
  const float* Wq = (const float*)d_in[5];
  const float* Wc = (const float*)d_in[6];
  const float* Wl = (const float*)d_in[7];
  const float* Wv = (const float*)d_in[8];
  const float* bv = (const float*)d_in[9];

  float* out      = (float*)d_out;
  float* ctx_out  = out;                          // (256, 512) = 131072
  float* attn_out = out + B_SZ * ENC_DIM;         // (256, 1024) = 262144

  float* ws       = (float*)d_ws;
  float* pq       = ws;                           // 32768 floats
  float* energies = ws + B_SZ * A_DIM;            // 262144 floats
  float* partial  = energies + B_SZ * L_SZ;       // 524288 floats (~3.2MB total)

  pq_wmma_kernel<<<dim3(A_DIM / 16, B_SZ / 16), 32, 0, stream>>>(query, Wq, pq);
  energies_kernel<<<dim3(L_SZ / 64, B_SZ), 256, 0, stream>>>(
      ps, attn_cat, mask, pq, Wc, Wl, Wv, bv, energies);
  softmax_kernel<<<B_SZ, 256, 0, stream>>>(energies, attn_out);
  context_partial_kernel<<<dim3(4, B_SZ), ENC_DIM, 0, stream>>>(attn_out, source, partial);
  context_reduce_kernel<<<(B_SZ * ENC_DIM + 255) / 256, 256, 0, stream>>>(partial, ctx_out);
}